// MBartAttention_9062380994925
// MI455X (gfx1250) — compile-verified
//
#include <hip/hip_runtime.h>

#define D_MODEL 1024
#define HD      64
#define NH      16
#define NB      16
#define SEQ     4096

typedef __attribute__((ext_vector_type(2))) float v2f;
typedef __attribute__((ext_vector_type(8))) float v8f;

// One wave computes a 16(M=batch) x 16(N) tile of  out = X @ W^T + bias, scaled.
// X: [16 x 1024] row-major, W: [1024 x 1024] row-major (PyTorch Linear weight),
// out: [16 x 1024]. Uses V_WMMA_F32_16X16X4_F32, K-loop of 256 steps.
__device__ __forceinline__ void gemm16_tile(const float* __restrict__ X,
                                            const float* __restrict__ W,
                                            const float* __restrict__ bias,
                                            float* __restrict__ out,
                                            int n0, float scale) {
  const int lane = threadIdx.x & 31;
  const int r    = lane & 15;          // M index for A, N index for B
  const int kk   = (lane >> 4) << 1;   // K sub-offset: lanes 0-15 -> K0,K1; 16-31 -> K2,K3
  v8f c = {};
  for (int ko = 0; ko < D_MODEL; ko += 4) {
    v2f a, b;
    a.x = X[r * D_MODEL + ko + kk];
    a.y = X[r * D_MODEL + ko + kk + 1];
    b.x = W[(n0 + r) * D_MODEL + ko + kk];       // B(k,n) = W[n][k]  (X @ W^T)
    b.y = W[(n0 + r) * D_MODEL + ko + kk + 1];
    c = __builtin_amdgcn_wmma_f32_16x16x4_f32(false, a, false, b, (short)0, c,
                                              false, false);
  }
  const int mbase = (lane >> 4) << 3;            // lanes 16-31 hold M=8..15
  const float bb = bias[n0 + r];
#pragma unroll
  for (int j = 0; j < 8; ++j) {
    out[(mbase + j) * D_MODEL + n0 + r] = (c[j] + bb) * scale;
  }
}

// blockIdx.x in [0,192): 64 N-tiles x {q,k,v}
__global__ void __launch_bounds__(32)
qkv_kernel(const float* __restrict__ hs,
           const float* __restrict__ Wq, const float* __restrict__ bq,
           const float* __restrict__ Wk, const float* __restrict__ bk,
           const float* __restrict__ Wv, const float* __restrict__ bv,
           float* __restrict__ q, float* __restrict__ k, float* __restrict__ v) {
  const int n0    = (blockIdx.x & 63) * 16;
  const int which = blockIdx.x >> 6;     // uniform per block -> EXEC stays all-ones
  if (which == 0)      gemm16_tile(hs, Wq, bq, q, n0, 0.125f);  // * hd^-0.5
  else if (which == 1) gemm16_tile(hs, Wk, bk, k, n0, 1.0f);
  else                 gemm16_tile(hs, Wv, bv, v, n0, 1.0f);
}

__global__ void __launch_bounds__(32)
oproj_kernel(const float* __restrict__ ao,
             const float* __restrict__ Wo, const float* __restrict__ bo,
             float* __restrict__ out) {
  gemm16_tile(ao, Wo, bo, out, blockIdx.x * 16, 1.0f);
}

// Scatter the new K/V slot into the (already memcpy'd) output caches.
__global__ void __launch_bounds__(256)
patch_kernel(const float* __restrict__ kn, const float* __restrict__ vn,
             float* __restrict__ kb, float* __restrict__ vb,
             const int* __restrict__ step_p) {
  const int idx  = blockIdx.x * blockDim.x + threadIdx.x;   // 0..16383
  const int step = *step_p;
  const int b  = idx >> 10;
  const int d  = idx & 1023;
  const int h  = d >> 6;
  const int hd = d & 63;
  const size_t off = (((size_t)(b * NH + h) * SEQ) + (size_t)step) * HD + hd;
  kb[off] = kn[idx];
  vb[off] = vn[idx];
}

// One workgroup per (b,h): scores over positions 0..step, softmax, P@V.
__global__ void __launch_bounds__(256)
attn_kernel(const float* __restrict__ q,
            const float* __restrict__ kb,
            const float* __restrict__ vb,
            float* __restrict__ out,
            const int* __restrict__ step_p) {
  __shared__ float qs[HD];
  __shared__ float sc[SEQ];
  __shared__ float red[256];
  __shared__ float vacc[4][HD];

  const int t    = threadIdx.x;
  const int bh   = blockIdx.x;          // b*NH + h
  const int step = *step_p;
  const int b = bh >> 4, h = bh & 15;
  const float* __restrict__ kbase = kb + (size_t)bh * SEQ * HD;
  const float* __restrict__ vbase = vb + (size_t)bh * SEQ * HD;

  if (t < HD) qs[t] = q[b * D_MODEL + h * HD + t];
  __syncthreads();

  // scores + running max
  float lmax = -3.4e38f;
  for (int s = t; s <= step; s += 256) {
    const float* __restrict__ kr = kbase + (size_t)s * HD;
    float dot = 0.f;
#pragma unroll
    for (int d2 = 0; d2 < HD; d2 += 4) {
      dot += qs[d2 + 0] * kr[d2 + 0] + qs[d2 + 1] * kr[d2 + 1] +
             qs[d2 + 2] * kr[d2 + 2] + qs[d2 + 3] * kr[d2 + 3];
    }
    sc[s] = dot;
    lmax = fmaxf(lmax, dot);
  }
  red[t] = lmax;
  __syncthreads();
  for (int off = 128; off > 0; off >>= 1) {
    if (t < off) red[t] = fmaxf(red[t], red[t + off]);
    __syncthreads();
  }
  const float gmax = red[0];
  __syncthreads();

  // exp + sum
  float lsum = 0.f;
  for (int s = t; s <= step; s += 256) {
    const float e = __expf(sc[s] - gmax);
    sc[s] = e;
    lsum += e;
  }
  red[t] = lsum;
  __syncthreads();
  for (int off = 128; off > 0; off >>= 1) {
    if (t < off) red[t] += red[t + off];
    __syncthreads();
  }
  const float gsum = red[0];
  __syncthreads();

  // P @ V: 4 position-groups x 64 dims -> coalesced V reads
  const int d = t & (HD - 1);
  const int g = t >> 6;
  float acc = 0.f;
  for (int s = g; s <= step; s += 4) {
    acc += sc[s] * vbase[(size_t)s * HD + d];
  }
  vacc[g][d] = acc;
  __syncthreads();
  if (g == 0) {
    const float tot = vacc[0][d] + vacc[1][d] + vacc[2][d] + vacc[3][d];
    out[b * D_MODEL + h * HD + d] = tot / gsum;
  }
}

extern "C" void kernel_launch(void* const* d_in, const int* in_sizes, int n_in,
                              void* d_out, int out_size, void* d_ws, size_t ws_size,
                              hipStream_t stream) {
  const float* hs   = (const float*)d_in[0];
  const float* keyb = (const float*)d_in[1];
  const float* valb = (const float*)d_in[2];
  const float* Wq   = (const float*)d_in[3];
  const float* bq   = (const float*)d_in[4];
  const float* Wk   = (const float*)d_in[5];
  const float* bk   = (const float*)d_in[6];
  const float* Wv   = (const float*)d_in[7];
  const float* bv   = (const float*)d_in[8];
  const float* Wo   = (const float*)d_in[9];
  const float* bo   = (const float*)d_in[10];
  const int*   step = (const int*)d_in[11];

  const size_t cacheElems = (size_t)NB * NH * SEQ * HD;   // 67,108,864
  float* out = (float*)d_out;                // [16*1024]
  float* kb  = out + (size_t)NB * D_MODEL;   // [B,H,S,HD]
  float* vb  = kb + cacheElems;

  float* q  = (float*)d_ws;                  // [16*1024]
  float* kn = q  + (size_t)NB * D_MODEL;
  float* vn = kn + (size_t)NB * D_MODEL;
  float* ao = vn + (size_t)NB * D_MODEL;

  // Bulk cache replication on the d2d copy path (~512 MB writes, the real cost).
  hipMemcpyAsync(kb, keyb, cacheElems * sizeof(float), hipMemcpyDeviceToDevice, stream);
  hipMemcpyAsync(vb, valb, cacheElems * sizeof(float), hipMemcpyDeviceToDevice, stream);

  qkv_kernel<<<192, 32, 0, stream>>>(hs, Wq, bq, Wk, bk, Wv, bv, q, kn, vn);
  patch_kernel<<<64, 256, 0, stream>>>(kn, vn, kb, vb, step);
  attn_kernel<<<NB * NH, 256, 0, stream>>>(q, kb, vb, ao, step);
  oproj_kernel<<<64, 32, 0, stream>>>(ao, Wo, bo, out);
}